// SparseNet_12489764897164
// MI455X (gfx1250) — compile-verified
//
#include <hip/hip_runtime.h>
#include <math.h>

typedef __attribute__((ext_vector_type(2))) float v2f;
typedef __attribute__((ext_vector_type(8))) float v8f;

// Problem constants (from reference)
#define SG    15      // spatial grid 15x15
#define KP    112     // sparse points per batch
#define CIN   200     // input channels
#define CHUNK 20      // channel chunk staged in LDS
#define NCHK  (CIN / CHUNK)
#define NEGF  (-1e30f)

__global__ __launch_bounds__(256) void sparsenet_kernel(
    const int*   __restrict__ coords,   // (B*KP, 3) = y,x,b
    const float* __restrict__ features, // (B*KP, 200)
    const float* __restrict__ W1,       // (3,3,200,16)
    const float* __restrict__ b1,       // (16)
    const float* __restrict__ W2,       // (5,5,16,32)
    const float* __restrict__ linW,     // (32,2)
    const float* __restrict__ linb,     // (2)
    float* __restrict__ out)            // (B,2)
{
    __shared__ float s_in[225 * CHUNK];      // dense input chunk  (18000 B)
    __shared__ float s_w[9 * CHUNK * 16];    // W1 chunk           (11520 B)
    __shared__ float s_h1[225 * 16];         // conv1 out, masked  (14400 B)
    __shared__ float s_mask[225];
    __shared__ float s_p1[49 * 16];          // pooled 7x7x16
    __shared__ float s_m1[49];
    __shared__ float s_h2[9 * 32];           // conv2 out 3x3x32
    __shared__ float s_m2[9];
    __shared__ float s_x[32];                // pooled 1x1x32

    const int b    = blockIdx.x;
    const int tid  = threadIdx.x;
    const int wave = tid >> 5;     // wave32: 8 waves per block
    const int lane = tid & 31;
    const int half = lane >> 4;    // lane half selects K={0,1} vs K={2,3}
    const int l16  = lane & 15;

    // ---- build occupancy mask ----
    for (int p = tid; p < 225; p += 256) s_mask[p] = 0.f;
    __syncthreads();
    for (int i = tid; i < KP; i += 256) {
        int r = b * KP + i;
        int y = coords[r * 3 + 0];
        int x = coords[r * 3 + 1];
        s_mask[y * SG + x] = 1.f;
    }

    // ---- conv1 as implicit GEMM: M=240(15 tiles), N=16, K=9*200 ----
    v8f acc0 = {};   // M-tile = wave
    v8f acc1 = {};   // M-tile = wave+8 (waves 0..6 only)

    for (int cc = 0; cc < NCHK; ++cc) {
        __syncthreads();  // previous chunk fully consumed before re-fill
        for (int e = tid; e < 225 * CHUNK; e += 256) s_in[e] = 0.f;
        for (int e = tid; e < 9 * CHUNK * 16; e += 256) {
            int tap = e / (CHUNK * 16);
            int rem = e - tap * (CHUNK * 16);
            int c  = rem >> 4;
            int oc = rem & 15;
            s_w[e] = W1[(tap * CIN + cc * CHUNK + c) * 16 + oc];
        }
        __syncthreads();
        for (int e = tid; e < KP * CHUNK; e += 256) {
            int i = e / CHUNK;
            int c = e - i * CHUNK;
            int r = b * KP + i;
            int y = coords[r * 3 + 0];
            int x = coords[r * 3 + 1];
            s_in[(y * SG + x) * CHUNK + c] = features[(size_t)r * CIN + cc * CHUNK + c];
        }
        __syncthreads();

        for (int tap = 0; tap < 9; ++tap) {
            int ky = tap / 3 - 1;
            int kx = tap - (tap / 3) * 3 - 1;
            #pragma unroll
            for (int k4 = 0; k4 < CHUNK / 4; ++k4) {
                int kb = k4 * 4 + half * 2;
                // B fragment: 4x16 W1 slice (K in lane-half, N=l16)
                v2f bf;
                bf.x = s_w[(tap * CHUNK + kb    ) * 16 + l16];
                bf.y = s_w[(tap * CHUNK + kb + 1) * 16 + l16];
                {   // tile 0 of this wave
                    int p  = wave * 16 + l16;
                    int py = p / SG, px = p - py * SG;
                    int iy = py + ky, ix = px + kx;
                    bool ok = (p < 225) & (iy >= 0) & (iy < SG) & (ix >= 0) & (ix < SG);
                    int ip = ok ? (iy * SG + ix) : 0;
                    v2f af;
                    af.x = s_in[ip * CHUNK + kb];
                    af.y = s_in[ip * CHUNK + kb + 1];
                    af.x = ok ? af.x : 0.f;
                    af.y = ok ? af.y : 0.f;
                    acc0 = __builtin_amdgcn_wmma_f32_16x16x4_f32(
                        false, af, false, bf, (short)0, acc0, false, false);
                }
                if (wave < 7) {  // wave-uniform branch: EXEC stays full
                    int p  = (8 + wave) * 16 + l16;
                    int py = p / SG, px = p - py * SG;
                    int iy = py + ky, ix = px + kx;
                    bool ok = (p < 225) & (iy >= 0) & (iy < SG) & (ix >= 0) & (ix < SG);
                    int ip = ok ? (iy * SG + ix) : 0;
                    v2f af;
                    af.x = s_in[ip * CHUNK + kb];
                    af.y = s_in[ip * CHUNK + kb + 1];
                    af.x = ok ? af.x : 0.f;
                    af.y = ok ? af.y : 0.f;
                    acc1 = __builtin_amdgcn_wmma_f32_16x16x4_f32(
                        false, af, false, bf, (short)0, acc1, false, false);
                }
            }
        }
    }

    // ---- write conv1 + bias, masked (C/D layout: VGPR r -> M = half*8 + r) ----
    {
        float bias = b1[l16];
        #pragma unroll
        for (int r = 0; r < 8; ++r) {
            int row = wave * 16 + half * 8 + r;
            float v = acc0[r] + bias;
            if (row < 225) s_h1[row * 16 + l16] = (s_mask[row] > 0.5f) ? v : 0.f;
        }
        if (wave < 7) {
            #pragma unroll
            for (int r = 0; r < 8; ++r) {
                int row = (8 + wave) * 16 + half * 8 + r;
                float v = acc1[r] + bias;
                if (row < 225) s_h1[row * 16 + l16] = (s_mask[row] > 0.5f) ? v : 0.f;
            }
        }
    }
    __syncthreads();

    // ---- masked maxpool 3x3 stride 2: 15x15 -> 7x7 ----
    for (int e = tid; e < 49 * 16; e += 256) {
        int oc   = e & 15;
        int opos = e >> 4;
        int oy = opos / 7, ox = opos - oy * 7;
        float mx = NEGF, mm = 0.f;
        #pragma unroll
        for (int dy = 0; dy < 3; ++dy)
            #pragma unroll
            for (int dx = 0; dx < 3; ++dx) {
                int p = (2 * oy + dy) * SG + (2 * ox + dx);
                float mk = s_mask[p];
                mm = fmaxf(mm, mk);
                float v = (mk > 0.5f) ? s_h1[p * 16 + oc] : NEGF;
                mx = fmaxf(mx, v);
            }
        s_p1[opos * 16 + oc] = (mm > 0.5f) ? mx : 0.f;
        if (oc == 0) s_m1[opos] = mm;
    }
    __syncthreads();

    // ---- conv2: 5x5 VALID, 16 -> 32, on 7x7 -> 3x3; masked by 5x5-any(m1) ----
    for (int e = tid; e < 9 * 32; e += 256) {
        int oc   = e & 31;
        int opos = e >> 5;
        int oy = opos / 3, ox = opos - oy * 3;
        float s = 0.f, mm = 0.f;
        for (int ky = 0; ky < 5; ++ky)
            for (int kx = 0; kx < 5; ++kx) {
                int ip = (oy + ky) * 7 + (ox + kx);
                mm = fmaxf(mm, s_m1[ip]);
                #pragma unroll
                for (int c = 0; c < 16; ++c)
                    s += s_p1[ip * 16 + c] * W2[((ky * 5 + kx) * 16 + c) * 32 + oc];
            }
        s_h2[opos * 32 + oc] = (mm > 0.5f) ? s : 0.f;
        if (oc == 0) s_m2[opos] = mm;
    }
    __syncthreads();

    // ---- masked maxpool 3x3 stride 2: 3x3 -> 1x1 ----
    if (tid < 32) {
        float mx = NEGF, mm = 0.f;
        #pragma unroll
        for (int p = 0; p < 9; ++p) {
            float mk = s_m2[p];
            mm = fmaxf(mm, mk);
            float v = (mk > 0.5f) ? s_h2[p * 32 + tid] : NEGF;
            mx = fmaxf(mx, v);
        }
        s_x[tid] = (mm > 0.5f) ? mx : 0.f;
    }
    __syncthreads();

    // ---- linear + softmax ----
    if (tid == 0) {
        float z0 = linb[0], z1 = linb[1];
        #pragma unroll
        for (int c = 0; c < 32; ++c) {
            z0 += s_x[c] * linW[c * 2 + 0];
            z1 += s_x[c] * linW[c * 2 + 1];
        }
        float m  = fmaxf(z0, z1);
        float e0 = expf(z0 - m);
        float e1 = expf(z1 - m);
        float inv = 1.f / (e0 + e1);
        out[b * 2 + 0] = e0 * inv;
        out[b * 2 + 1] = e1 * inv;
    }
}

extern "C" void kernel_launch(void* const* d_in, const int* in_sizes, int n_in,
                              void* d_out, int out_size, void* d_ws, size_t ws_size,
                              hipStream_t stream) {
    const int*   coords   = (const int*)  d_in[0];
    const float* features = (const float*)d_in[1];
    const float* W1       = (const float*)d_in[2];
    const float* b1       = (const float*)d_in[3];
    const float* W2       = (const float*)d_in[4];
    const float* linW     = (const float*)d_in[5];
    const float* linb     = (const float*)d_in[6];
    // d_in[7] is the device-side batch_size scalar; derive B host-side instead:
    int B = in_sizes[0] / (KP * 3);
    float* out = (float*)d_out;

    sparsenet_kernel<<<B, 256, 0, stream>>>(coords, features, W1, b1, W2,
                                            linW, linb, out);
}